// eGAT_65171833750213
// MI455X (gfx1250) — compile-verified
//
#include <hip/hip_runtime.h>
#include <math.h>

typedef float v2f __attribute__((ext_vector_type(2)));
typedef float v8f __attribute__((ext_vector_type(8)));

#define NEG_SLOPE 0.2f

// ---------------------------------------------------------------------------
// utility: order-preserving float atomic max (safe without float atomicMax)
// ---------------------------------------------------------------------------
__device__ __forceinline__ void atomicMaxF(float* addr, float val) {
    if (val >= 0.0f)
        atomicMax((int*)addr, __float_as_int(val));
    else
        atomicMin((unsigned int*)addr, (unsigned int)__float_as_int(val));
}

// ---------------------------------------------------------------------------
// fill buffer with constant
// ---------------------------------------------------------------------------
__global__ void egat_fill(float* __restrict__ p, float v, long long n) {
    long long i = (long long)blockIdx.x * blockDim.x + threadIdx.x;
    if (i < n) p[i] = v;
}

// ---------------------------------------------------------------------------
// C[M,128] = A[M,128] @ B[128,128] (+bias) (+relu)   -- fp32 WMMA 16x16x4
// 256 threads = 8 waves; each wave owns a 16-row x 128-col strip.
//
// B staged in LDS in FRAGMENT-MAJOR float2 layout:
//   Bs2[kp*128 + (c ^ ((kp&1)<<4))] = { B[2kp][c], B[2kp+1][c] }
// so each WMMA B-fragment is one aligned ds_load_b64 (no repack movs).
// The (kp&1)<<4 column swizzle makes the two 16-lane halves (which read
// adjacent kp rows, since kp = k/2 + halfk) hit complementary bank sets.
//
// A fragments are read straight from global, clamped to a valid row for
// out-of-range lanes (row m of C depends only on row m of A; those rows are
// dropped at the store guard) -> no EXEC-mask branches around VMEM loads.
// ---------------------------------------------------------------------------
__global__ __launch_bounds__(256)
void egat_gemm128_wmma(const float* __restrict__ A, const float* __restrict__ B,
                       const float* __restrict__ bias, float* __restrict__ C,
                       int M, int relu) {
    __shared__ v2f Bs2[64 * 128];   // 64KB
    const int tid    = threadIdx.x;
    const int waveId = tid >> 5;
    const int lane   = tid & 31;
    const int lm     = lane & 15;   // column / row-in-tile index
    const int halfk  = lane >> 4;   // which K-pair this lane holds
    const int row0   = blockIdx.x * 128;

    // stage B: pack row pairs into float2 fragments with bank swizzle
    for (int u = tid; u < 64 * 32; u += 256) {
        const int rp = u >> 5;          // k-pair row (0..63)
        const int cg = u & 31;          // column group of 4 (0..31)
        const float4 a0 = *(const float4*)&B[(size_t)(rp * 2    ) * 128 + cg * 4];
        const float4 a1 = *(const float4*)&B[(size_t)(rp * 2 + 1) * 128 + cg * 4];
        const int cs = (cg * 4) ^ ((rp & 1) << 4);
        v2f* dst = &Bs2[rp * 128 + cs];
        v2f t0; t0.x = a0.x; t0.y = a1.x; dst[0] = t0;
        v2f t1; t1.x = a0.y; t1.y = a1.y; dst[1] = t1;
        v2f t2; t2.x = a0.z; t2.y = a1.z; dst[2] = t2;
        v2f t3; t3.x = a0.w; t3.y = a1.w; dst[3] = t3;
    }
    __syncthreads();

    const int m0  = waveId * 16;
    const int gmA = row0 + m0 + lm;                  // A row this lane supplies
    const int gmC = gmA < M ? gmA : (M - 1);         // clamped, always valid
    const float* Arow = A + (size_t)gmC * 128;

    v8f acc[8] = {};
    for (int k = 0; k < 128; k += 4) {
        const int kp = (k >> 1) + halfk;             // kp & 1 == halfk
        v2f a = *(const v2f*)(Arow + k + halfk * 2); // one global_load_b64
#pragma unroll
        for (int t = 0; t < 8; t++) {
            v2f b = Bs2[kp * 128 + ((t ^ halfk) * 16 + lm)];  // one ds_load_b64
            acc[t] = __builtin_amdgcn_wmma_f32_16x16x4_f32(
                false, a, false, b, (short)0, acc[t], false, false);
        }
    }

    const bool fullTile = (row0 + 128 <= M);         // uniform fast path
#pragma unroll
    for (int t = 0; t < 8; t++) {
        const int   n  = t * 16 + lm;
        const float bv = bias ? bias[n] : 0.0f;
        float* Cp = C + (size_t)(row0 + m0 + halfk * 8) * 128 + n;
#pragma unroll
        for (int r = 0; r < 8; r++) {
            float v = acc[t][r] + bv;
            if (relu) v = fmaxf(v, 0.0f);
            if (fullTile || (row0 + m0 + halfk * 8 + r) < M)
                Cp[(size_t)r * 128] = v;
        }
    }
}

// ---------------------------------------------------------------------------
// per-node attention logits: alpha_src[i] = xl[i]·a_s ; alpha_dst[i] = xl[i]·a_d
// one wave per node, float4 per lane, wave reduction.
// ---------------------------------------------------------------------------
__global__ void egat_node_alpha(const float* __restrict__ xl,
                                const float* __restrict__ a_s,
                                const float* __restrict__ a_d,
                                float* __restrict__ alpha_src,
                                float* __restrict__ alpha_dst, int N) {
    int gw   = (blockIdx.x * blockDim.x + threadIdx.x) >> 5;
    int lane = threadIdx.x & 31;
    if (gw >= N) return;
    float4 v = *(const float4*)&xl[(size_t)gw * 128 + lane * 4];
    float4 s = *(const float4*)&a_s[lane * 4];
    float4 d = *(const float4*)&a_d[lane * 4];
    float ps = v.x * s.x + v.y * s.y + v.z * s.z + v.w * s.w;
    float pd = v.x * d.x + v.y * d.y + v.z * d.z + v.w * d.w;
    for (int off = 16; off; off >>= 1) {
        ps += __shfl_down(ps, off);
        pd += __shfl_down(pd, off);
    }
    if (lane == 0) { alpha_src[gw] = ps; alpha_dst[gw] = pd; }
}

// ---------------------------------------------------------------------------
// q[k] = We[k,:] · a_e   (16 waves, one per k)
// ---------------------------------------------------------------------------
__global__ void egat_we_ae(const float* __restrict__ We,
                           const float* __restrict__ a_e, float* __restrict__ q) {
    int k = blockIdx.x, lane = threadIdx.x;
    float4 w = *(const float4*)&We[(size_t)k * 128 + lane * 4];
    float4 a = *(const float4*)&a_e[lane * 4];
    float p = w.x * a.x + w.y * a.y + w.z * a.z + w.w * a.w;
    for (int off = 16; off; off >>= 1) p += __shfl_down(p, off);
    if (lane == 0) q[k] = p;
}

// ---------------------------------------------------------------------------
// degree + edge_attr segment-sum (for self-loop fill 'mean')
// one thread per (edge, attr) pair
// ---------------------------------------------------------------------------
__global__ void egat_edge_mean_accum(const long long* __restrict__ ei,
                                     const float* __restrict__ ea,
                                     float* __restrict__ deg,
                                     float* __restrict__ ea_sum, int E) {
    int idx = blockIdx.x * blockDim.x + threadIdx.x;
    if (idx >= E * 16) return;
    int e = idx >> 4, k = idx & 15;
    int dst = (int)ei[(size_t)E + e];
    atomicAdd(&ea_sum[(size_t)dst * 16 + k], ea[idx]);
    if (k == 0) atomicAdd(&deg[dst], 1.0f);
}

__global__ void egat_edge_mean_final(float* __restrict__ ea_mean,
                                     const float* __restrict__ deg, int N) {
    int idx = blockIdx.x * blockDim.x + threadIdx.x;
    if (idx >= N * 16) return;
    ea_mean[idx] = ea_mean[idx] / fmaxf(deg[idx >> 4], 1.0f);
}

// ---------------------------------------------------------------------------
// edge pass 1: alpha = leaky_relu(asrc[s] + adst[d] + ea·q); segment atomic max
// edges [0,E) are real edges; [E, E+N) are self loops (use ea_mean)
// ---------------------------------------------------------------------------
__global__ void egat_edge_alpha(const long long* __restrict__ ei,
                                const float* __restrict__ ea,
                                const float* __restrict__ ea_mean,
                                const float* __restrict__ asrc,
                                const float* __restrict__ adst,
                                const float* __restrict__ q,
                                float* __restrict__ alpha,
                                float* __restrict__ amax, int E, int N) {
    int e = blockIdx.x * blockDim.x + threadIdx.x;
    if (e >= E + N) return;
    int s, d;
    const float* ep;
    if (e < E) { s = (int)ei[e]; d = (int)ei[(size_t)E + e]; ep = &ea[(size_t)e * 16]; }
    else       { s = d = e - E;                               ep = &ea_mean[(size_t)(e - E) * 16]; }
    float ae = 0.0f;
#pragma unroll
    for (int k = 0; k < 16; k++) ae += ep[k] * q[k];
    float a = asrc[s] + adst[d] + ae;
    a = (a > 0.0f) ? a : NEG_SLOPE * a;
    alpha[e] = a;
    atomicMaxF(&amax[d], a);
}

// edge pass 2: ex = exp(alpha - amax[dst]); denom[dst] += ex
__global__ void egat_edge_exp(const long long* __restrict__ ei,
                              float* __restrict__ alpha,
                              const float* __restrict__ amax,
                              float* __restrict__ denom, int E, int N) {
    int e = blockIdx.x * blockDim.x + threadIdx.x;
    if (e >= E + N) return;
    int d = (e < E) ? (int)ei[(size_t)E + e] : (e - E);
    float ex = __expf(alpha[e] - amax[d]);
    alpha[e] = ex;
    atomicAdd(&denom[d], ex);
}

// edge pass 3: agg[dst] += xl[src] * (ex/denom[dst]); one wave per edge
__global__ void egat_edge_aggregate(const long long* __restrict__ ei,
                                    const float* __restrict__ xl,
                                    const float* __restrict__ ex,
                                    const float* __restrict__ denom,
                                    float* __restrict__ agg, int E, int N) {
    int gw   = (blockIdx.x * blockDim.x + threadIdx.x) >> 5;
    int lane = threadIdx.x & 31;
    if (gw >= E + N) return;
    int s, d;
    if (gw < E) { s = (int)ei[gw]; d = (int)ei[(size_t)E + gw]; }
    else        { s = d = gw - E; }
    float w = ex[gw] / denom[d];
    float4 v = *(const float4*)&xl[(size_t)s * 128 + lane * 4];
    float* o = &agg[(size_t)d * 128 + lane * 4];
    atomicAdd(o + 0, v.x * w);
    atomicAdd(o + 1, v.y * w);
    atomicAdd(o + 2, v.z * w);
    atomicAdd(o + 3, v.w * w);
}

// h = relu(agg + b)
__global__ void egat_bias_relu(float* __restrict__ h, const float* __restrict__ b, int N) {
    int idx = blockIdx.x * blockDim.x + threadIdx.x;
    if (idx >= N * 128) return;
    h[idx] = fmaxf(h[idx] + b[idx & 127], 0.0f);
}

// ---------------------------------------------------------------------------
// logits = h @ lin2_w + lin2_b ; log_softmax over 16 classes
// one wave per node: lanes split 128-d dot in halves, classes = lane&15
// ---------------------------------------------------------------------------
__global__ void egat_final(const float* __restrict__ h, const float* __restrict__ W2,
                           const float* __restrict__ b2, float* __restrict__ out, int N) {
    int gw   = (blockIdx.x * blockDim.x + threadIdx.x) >> 5;
    int lane = threadIdx.x & 31;
    if (gw >= N) return;
    int c = lane & 15, half = lane >> 4;
    const float* hp = &h[(size_t)gw * 128 + half * 64];
    float z = 0.0f;
    for (int j = 0; j < 64; j++) z += hp[j] * W2[(half * 64 + j) * 16 + c];
    z += __shfl_down(z, 16);          // lanes 0..15 now hold full dot
    z += b2[c];
    float m = z;
    for (int off = 8; off; off >>= 1) m = fmaxf(m, __shfl_xor(m, off, 16));
    float eo = __expf(z - m), sum = eo;
    for (int off = 8; off; off >>= 1) sum += __shfl_xor(sum, off, 16);
    if (lane < 16) out[(size_t)gw * 16 + c] = (z - m) - __logf(sum);
}

// ---------------------------------------------------------------------------
// host side
// ---------------------------------------------------------------------------
static void run_gat_layer(const float* h_in, const long long* ei,
                          const float* ea, const float* ea_mean,
                          const float* W, const float* a_s, const float* a_d,
                          const float* We, const float* a_e, const float* b,
                          float* xl, float* asrc, float* adst, float* amax,
                          float* denom, float* alpha, float* q, float* h_out,
                          int N, int E, hipStream_t stream) {
    const long long NF = (long long)N * 128;
    egat_fill<<<(int)((NF + 255) / 256), 256, 0, stream>>>(h_out, 0.0f, NF);
    egat_fill<<<(N + 255) / 256, 256, 0, stream>>>(amax, -INFINITY, N);
    egat_fill<<<(N + 255) / 256, 256, 0, stream>>>(denom, 0.0f, N);

    egat_gemm128_wmma<<<(N + 127) / 128, 256, 0, stream>>>(h_in, W, nullptr, xl, N, 0);
    egat_node_alpha<<<(N * 32 + 255) / 256, 256, 0, stream>>>(xl, a_s, a_d, asrc, adst, N);
    egat_we_ae<<<16, 32, 0, stream>>>(We, a_e, q);

    int EN = E + N;
    egat_edge_alpha<<<(EN + 255) / 256, 256, 0, stream>>>(ei, ea, ea_mean, asrc, adst,
                                                          q, alpha, amax, E, N);
    egat_edge_exp<<<(EN + 255) / 256, 256, 0, stream>>>(ei, alpha, amax, denom, E, N);
    egat_edge_aggregate<<<(int)(((long long)EN * 32 + 255) / 256), 256, 0, stream>>>(
        ei, xl, alpha, denom, h_out, E, N);
    egat_bias_relu<<<(int)((NF + 255) / 256), 256, 0, stream>>>(h_out, b, N);
}

extern "C" void kernel_launch(void* const* d_in, const int* in_sizes, int n_in,
                              void* d_out, int out_size, void* d_ws, size_t ws_size,
                              hipStream_t stream) {
    (void)n_in; (void)out_size; (void)ws_size;
    const float*     x      = (const float*)d_in[0];
    const float*     ea     = (const float*)d_in[1];
    const float*     W0     = (const float*)d_in[2];
    const float*     a_s0   = (const float*)d_in[3];
    const float*     a_d0   = (const float*)d_in[4];
    const float*     We0    = (const float*)d_in[5];
    const float*     a_e0   = (const float*)d_in[6];
    const float*     b0     = (const float*)d_in[7];
    const float*     W1     = (const float*)d_in[8];
    const float*     a_s1   = (const float*)d_in[9];
    const float*     a_d1   = (const float*)d_in[10];
    const float*     We1    = (const float*)d_in[11];
    const float*     a_e1   = (const float*)d_in[12];
    const float*     b1     = (const float*)d_in[13];
    const float*     lin1_w = (const float*)d_in[14];
    const float*     lin1_b = (const float*)d_in[15];
    const float*     lin2_w = (const float*)d_in[16];
    const float*     lin2_b = (const float*)d_in[17];
    const long long* ei     = (const long long*)d_in[18];
    float* out = (float*)d_out;

    const int N = in_sizes[0] / 128;   // 100000
    const int E = in_sizes[1] / 16;    // 1600000

    // workspace layout (all f32)
    float* ws      = (float*)d_ws;
    float* xl      = ws;                        // N*128
    float* bufA    = xl      + (size_t)N * 128; // N*128
    float* bufB    = bufA    + (size_t)N * 128; // N*128
    float* asrc    = bufB    + (size_t)N * 128; // N
    float* adst    = asrc    + N;               // N
    float* deg     = adst    + N;               // N
    float* amax    = deg     + N;               // N
    float* denom   = amax    + N;               // N
    float* ea_mean = denom   + N;               // N*16
    float* alpha   = ea_mean + (size_t)N * 16;  // E+N
    float* q       = alpha   + ((size_t)E + N); // 16

    // ---- self-loop edge_attr mean (shared by both layers) ----
    egat_fill<<<(N + 255) / 256, 256, 0, stream>>>(deg, 0.0f, N);
    egat_fill<<<(N * 16 + 255) / 256, 256, 0, stream>>>(ea_mean, 0.0f, (long long)N * 16);
    egat_edge_mean_accum<<<(E * 16 + 255) / 256, 256, 0, stream>>>(ei, ea, deg, ea_mean, E);
    egat_edge_mean_final<<<(N * 16 + 255) / 256, 256, 0, stream>>>(ea_mean, deg, N);

    // ---- layer 1: x -> bufA ----
    run_gat_layer(x, ei, ea, ea_mean, W0, a_s0, a_d0, We0, a_e0, b0,
                  xl, asrc, adst, amax, denom, alpha, q, bufA, N, E, stream);
    // ---- layer 2: bufA -> bufB ----
    run_gat_layer(bufA, ei, ea, ea_mean, W1, a_s1, a_d1, We1, a_e1, b1,
                  xl, asrc, adst, amax, denom, alpha, q, bufB, N, E, stream);

    // ---- MLP head: bufB -> bufA (relu(lin1)) -> out (log_softmax(lin2)) ----
    egat_gemm128_wmma<<<(N + 127) / 128, 256, 0, stream>>>(bufB, lin1_w, lin1_b, bufA, N, 1);
    egat_final<<<(N * 32 + 255) / 256, 256, 0, stream>>>(bufA, lin2_w, lin2_b, out, N);
}